// GCN_75617194213390
// MI455X (gfx1250) — compile-verified
//
#include <hip/hip_runtime.h>
#include <hip/hip_bf16.h>
#include <math.h>

#define N_NODES 100000
#define N_EDGES 1600000
#define IN_DIM  128
#define HIDDEN  128
#define OUT_DIM 64

typedef __attribute__((ext_vector_type(2))) float v2f;
typedef __attribute__((ext_vector_type(8))) float v8f;

// ---------------------------------------------------------------------------
// Integer in-degree count: deg[col[e]] += 1
// ---------------------------------------------------------------------------
__global__ __launch_bounds__(256) void count_kernel(const int* __restrict__ col,
                                                    int* __restrict__ deg, int E) {
    int e = blockIdx.x * blockDim.x + threadIdx.x;
    if (e < E) atomicAdd(&deg[col[e]], 1);
}

// dinv[i] = deg>0 ? rsqrt(deg) : 0
__global__ __launch_bounds__(256) void dinv_kernel(const int* __restrict__ deg,
                                                   float* __restrict__ dinv, int n) {
    int i = blockIdx.x * blockDim.x + threadIdx.x;
    if (i < n) {
        int d = deg[i];
        dinv[i] = (d > 0) ? rsqrtf((float)d) : 0.0f;
    }
}

// ---------------------------------------------------------------------------
// Exclusive prefix sum of degrees -> CSR row offsets.
// Single block, 1024 threads, sequential 1024-wide chunks (Hillis-Steele in LDS).
// ---------------------------------------------------------------------------
__global__ __launch_bounds__(1024) void scan_kernel(const int* __restrict__ deg,
                                                    int* __restrict__ offs, int n) {
    __shared__ int smem[1024];
    __shared__ int running;
    const int tid = threadIdx.x;
    if (tid == 0) running = 0;
    __syncthreads();
    for (int base = 0; base < n; base += 1024) {
        int v = (base + tid < n) ? deg[base + tid] : 0;
        smem[tid] = v;
        __syncthreads();
        #pragma unroll
        for (int off = 1; off < 1024; off <<= 1) {
            int t = (tid >= off) ? smem[tid - off] : 0;
            __syncthreads();
            smem[tid] += t;
            __syncthreads();
        }
        const int incl  = smem[tid];
        const int total = smem[1023];
        if (base + tid < n) offs[base + tid] = running + incl - v;  // exclusive
        __syncthreads();
        if (tid == 0) running += total;
        __syncthreads();
    }
}

// ---------------------------------------------------------------------------
// CSR bucket fill: for each edge, place (src=row, w=dinv[row]*dinv[col]) into
// destination col's bucket. 1.6M int atomics total (vs 300M float atomics for
// direct scatter-add of features).
// ---------------------------------------------------------------------------
__global__ __launch_bounds__(256) void fill_kernel(const int* __restrict__ row,
                                                   const int* __restrict__ col,
                                                   const int* __restrict__ offs,
                                                   int* __restrict__ cursor,
                                                   const float* __restrict__ dinv,
                                                   int* __restrict__ esrc,
                                                   float* __restrict__ ew, int E) {
    int e = blockIdx.x * blockDim.x + threadIdx.x;
    if (e >= E) return;
    const int r = row[e];
    const int c = col[e];
    const int p = offs[c] + atomicAdd(&cursor[c], 1);
    esrc[p] = r;
    ew[p]   = dinv[r] * dinv[c];
}

// ---------------------------------------------------------------------------
// fp32 GEMM via V_WMMA_F32_16X16X4_F32. One 16x16 C-tile per wave32.
// (codegen-verified: emits v_wmma_f32_16x16x4_f32 with pipelined loads)
// ---------------------------------------------------------------------------
template<int K, int N>
__global__ __launch_bounds__(256) void gemm_wmma_f32(const float* __restrict__ A,
                                                     const float* __restrict__ B,
                                                     float* __restrict__ C, int M) {
    const int lane    = threadIdx.x & 31;
    const int wave    = threadIdx.x >> 5;
    const int tilesN  = N / 16;
    const int t       = blockIdx.x * (blockDim.x >> 5) + wave;
    const int nTiles  = (M / 16) * tilesN;
    if (t >= nTiles) return;                 // wave-uniform: EXEC stays all-ones

    const int rowTile = t / tilesN;
    const int colTile = t % tilesN;
    const int m       = rowTile * 16 + (lane & 15);
    const int n       = colTile * 16 + (lane & 15);
    const int khalf   = (lane >> 4) * 2;     // 0 or 2

    v8f acc = {};
    #pragma unroll 8
    for (int k = 0; k < K; k += 4) {
        v2f a, b;
        a.x = A[(size_t)m * K + k + khalf];
        a.y = A[(size_t)m * K + k + khalf + 1];
        b.x = B[(size_t)(k + khalf)     * N + n];
        b.y = B[(size_t)(k + khalf + 1) * N + n];
        acc = __builtin_amdgcn_wmma_f32_16x16x4_f32(
                  false, a, false, b, (short)0, acc, false, false);
    }

    const int mBase = rowTile * 16 + ((lane >> 4) * 8);
    #pragma unroll
    for (int v = 0; v < 8; ++v)
        C[(size_t)(mBase + v) * N + n] = acc[v];
}

// ---------------------------------------------------------------------------
// CSR gather-aggregate: out[n,:] = act( sum_{j in bucket(n)} xw[esrc[j],:]*ew[j] + b )
// One wave per node; lane owns D/32 contiguous floats; register accumulation;
// each output row written exactly once (no atomics, no accumulator zero-fill).
// ---------------------------------------------------------------------------
template<int D, bool RELU>
__global__ __launch_bounds__(256) void gather_aggregate_kernel(
        const float* __restrict__ xw, const int* __restrict__ offs,
        const int* __restrict__ esrc, const float* __restrict__ ew,
        const float* __restrict__ bias, float* __restrict__ out,
        int nNodes, int nEdges) {
    constexpr int VEC = D / 32;
    const int lane = threadIdx.x & 31;
    const int node = (int)(((size_t)blockIdx.x * blockDim.x + threadIdx.x) >> 5);
    if (node >= nNodes) return;
    const int start = offs[node];
    const int end   = (node + 1 < nNodes) ? offs[node + 1] : nEdges;
    const int d0    = lane * VEC;

    float acc[VEC];
    #pragma unroll
    for (int i = 0; i < VEC; ++i) acc[i] = 0.0f;

    #pragma unroll 2
    for (int j = start; j < end; ++j) {
        const int   r = esrc[j];                       // uniform across wave
        const float w = ew[j];
        const float* __restrict__ src = xw + (size_t)r * D + d0;
        if constexpr (VEC == 4) {
            float4 v = *(const float4*)src;
            acc[0] += v.x * w; acc[1] += v.y * w;
            acc[2] += v.z * w; acc[3] += v.w * w;
        } else {
            float2 v = *(const float2*)src;
            acc[0] += v.x * w; acc[1] += v.y * w;
        }
    }

    float* __restrict__ dst = out + (size_t)node * D + d0;
    #pragma unroll
    for (int i = 0; i < VEC; ++i) {
        float v = acc[i] + bias[d0 + i];
        if (RELU) v = fmaxf(v, 0.0f);
        dst[i] = v;
    }
}

// ---------------------------------------------------------------------------
// Edge score: sigmoid( dot(h[row], h[col]) ) over OUT_DIM=64.
// One wave per edge, float2 per lane, shfl_xor tree reduction.
// ---------------------------------------------------------------------------
__global__ __launch_bounds__(256) void edge_score_kernel(const float* __restrict__ h,
                                                         const int* __restrict__ row,
                                                         const int* __restrict__ col,
                                                         float* __restrict__ out, int E) {
    const int lane = threadIdx.x & 31;
    const int e = (int)(((size_t)blockIdx.x * blockDim.x + threadIdx.x) >> 5);
    if (e >= E) return;
    const int r = row[e];
    const int c = col[e];
    float2 a = *(const float2*)(h + (size_t)r * OUT_DIM + lane * 2);
    float2 b = *(const float2*)(h + (size_t)c * OUT_DIM + lane * 2);
    float p = a.x * b.x + a.y * b.y;
    #pragma unroll
    for (int m = 16; m >= 1; m >>= 1) p += __shfl_xor(p, m, 32);
    if (lane == 0) out[e] = 1.0f / (1.0f + expf(-p));
}

// ---------------------------------------------------------------------------
extern "C" void kernel_launch(void* const* d_in, const int* in_sizes, int n_in,
                              void* d_out, int out_size, void* d_ws, size_t ws_size,
                              hipStream_t stream) {
    const float* x  = (const float*)d_in[0];
    const int*   ei = (const int*)  d_in[1];   // shape (2, E) flattened
    const float* W1 = (const float*)d_in[2];
    const float* b1 = (const float*)d_in[3];
    const float* W2 = (const float*)d_in[4];
    const float* b2 = (const float*)d_in[5];
    float* out = (float*)d_out;

    const int* row = ei;
    const int* col = ei + N_EDGES;

    // Workspace layout in bytes, 256-byte aligned sections.
    #define ALIGN256(v) ((((size_t)(v)) + 255) & ~(size_t)255)
    char* base = (char*)d_ws;
    size_t off = 0;
    int*   deg_i  = (int*)  (base + off); off = ALIGN256(off + (size_t)N_NODES * 4);
    float* dinv   = (float*)(base + off); off = ALIGN256(off + (size_t)N_NODES * 4);
    int*   offs   = (int*)  (base + off); off = ALIGN256(off + (size_t)N_NODES * 4);
    int*   cursor = (int*)  (base + off); off = ALIGN256(off + (size_t)N_NODES * 4);
    int*   esrc   = (int*)  (base + off); off = ALIGN256(off + (size_t)N_EDGES * 4);
    float* ew     = (float*)(base + off); off = ALIGN256(off + (size_t)N_EDGES * 4);
    float* xw1    = (float*)(base + off); off = ALIGN256(off + (size_t)N_NODES * HIDDEN * 4); // reused as hw2
    float* h      = (float*)(base + off); off = ALIGN256(off + (size_t)N_NODES * HIDDEN * 4);
    float* h2     = (float*)(base + off); off = ALIGN256(off + (size_t)N_NODES * OUT_DIM * 4);
    (void)ws_size; (void)in_sizes; (void)n_in; (void)out_size;

    hipMemsetAsync(deg_i,  0, (size_t)N_NODES * sizeof(int), stream);
    hipMemsetAsync(cursor, 0, (size_t)N_NODES * sizeof(int), stream);

    // --- Graph preprocessing: degrees, normalization, CSR by destination ----
    count_kernel<<<(N_EDGES + 255) / 256, 256, 0, stream>>>(col, deg_i, N_EDGES);
    dinv_kernel <<<(N_NODES + 255) / 256, 256, 0, stream>>>(deg_i, dinv, N_NODES);
    scan_kernel <<<1, 1024, 0, stream>>>(deg_i, offs, N_NODES);
    fill_kernel <<<(N_EDGES + 255) / 256, 256, 0, stream>>>(row, col, offs, cursor,
                                                            dinv, esrc, ew, N_EDGES);

    // --- Layer 1: xw1 = x @ W1 ; h = relu(gather_agg + b1) -----------------
    gemm_wmma_f32<IN_DIM, HIDDEN><<<(N_NODES / 16) * (HIDDEN / 16) / 8, 256, 0, stream>>>(
        x, W1, xw1, N_NODES);
    gather_aggregate_kernel<HIDDEN, true><<<(N_NODES * 32 + 255) / 256, 256, 0, stream>>>(
        xw1, offs, esrc, ew, b1, h, N_NODES, N_EDGES);

    // --- Layer 2: hw2 (in xw1) = h @ W2 ; h2 = gather_agg + b2 -------------
    gemm_wmma_f32<HIDDEN, OUT_DIM><<<(N_NODES / 16) * (OUT_DIM / 16) / 8, 256, 0, stream>>>(
        h, W2, xw1, N_NODES);
    gather_aggregate_kernel<OUT_DIM, false><<<(N_NODES * 32 + 255) / 256, 256, 0, stream>>>(
        xw1, offs, esrc, ew, b2, h2, N_NODES, N_EDGES);

    // --- Edge scores --------------------------------------------------------
    edge_score_kernel<<<(N_EDGES * 32) / 256, 256, 0, stream>>>(h2, row, col, out, N_EDGES);
}